// SpikeRoutedCorrelation_48112223649982
// MI455X (gfx1250) — compile-verified
//
#include <hip/hip_runtime.h>

typedef float v2f __attribute__((ext_vector_type(2)));
typedef float v4f __attribute__((ext_vector_type(4)));
typedef float v8f __attribute__((ext_vector_type(8)));

#define BATCH 4
#define CH    96
#define H_    192
#define W_    320
#define HW    (H_ * W_)        // 61440
#define OUTC  81

// One wave (32 lanes) handles 64 consecutive pixels of one image.
// lane = 32 lanes; col = lane&15 selects a group of 4 pixels (float4);
// half = lane>>4 selects channel classes {0,1} (half=0) or {2,3} (half=1),
// matching the V_WMMA_F32_16X16X4_F32 B-matrix layout:
//   B VGPR j, lanes 0-15  -> (K=j,   N=lane)
//   B VGPR j, lanes 16-31 -> (K=j+2, N=lane-16)
// With A = all-ones, D[m][n] = sum_k B[k][n]: the matrix pipe performs the
// 4-way K-class reduction; D VGPR0 leaves lane l holding the full sum for
// pixel column (l&15) -- exactly the lane that must store it.
__global__ __launch_bounds__(256) void
spike_corr_kernel(const float* __restrict__ f1,
                  const float* __restrict__ f2,
                  float* __restrict__ out) {
  const int lane = threadIdx.x & 31;
  const int wid  = (blockIdx.x * blockDim.x + threadIdx.x) >> 5;
  const int col  = lane & 15;
  const int half = lane >> 4;

  const int base = wid * 64;                 // first pixel of this wave's tile
  const int b    = base / HW;
  const int hw   = base - b * HW;            // multiple of 64 -> 16B aligned

  const float* p1 = f1 + (size_t)b * CH * HW + hw + 4 * col;
  const float* p2 = f2 + (size_t)b * CH * HW + hw + 4 * col;

  // Accumulators: component g (0..3) = pixel group; j = class within half.
  v4f dotj0 = {0.f, 0.f, 0.f, 0.f};   // class 2*half + 0
  v4f dotj1 = {0.f, 0.f, 0.f, 0.f};   // class 2*half + 1
  v4f absj0 = {0.f, 0.f, 0.f, 0.f};
  v4f absj1 = {0.f, 0.f, 0.f, 0.f};

  // 96 channels, 4 per step (one per class). Non-temporal: single-use stream.
#pragma unroll 4
  for (int t = 0; t < CH / 4; ++t) {
    const size_t o0 = (size_t)(4 * t + 2 * half + 0) * HW;
    const size_t o1 = (size_t)(4 * t + 2 * half + 1) * HW;
    v4f a0 = __builtin_nontemporal_load((const v4f*)(p1 + o0));
    v4f a1 = __builtin_nontemporal_load((const v4f*)(p1 + o1));
    v4f b0 = __builtin_nontemporal_load((const v4f*)(p2 + o0));
    v4f b1 = __builtin_nontemporal_load((const v4f*)(p2 + o1));
    dotj0 += a0 * b0;
    dotj1 += a1 * b1;
    absj0 += __builtin_elementwise_abs(a0);
    absj1 += __builtin_elementwise_abs(a1);
  }

  const v2f ones = {1.0f, 1.0f};
  const v8f zero = {0.f, 0.f, 0.f, 0.f, 0.f, 0.f, 0.f, 0.f};

  float corr[4];
#pragma unroll
  for (int g = 0; g < 4; ++g) {
    v2f bd = {dotj0[g], dotj1[g]};   // VGPR j -> K class (2*half + j)
    v2f ba = {absj0[g], absj1[g]};
    // D[m][n] = sum_k bd[k][n] : full channel dot for pixel hw + 4n + g.
    v8f D = __builtin_amdgcn_wmma_f32_16x16x4_f32(
        false, ones, false, bd, (short)0, zero, false, false);
    v8f A = __builtin_amdgcn_wmma_f32_16x16x4_f32(
        false, ones, false, ba, (short)0, zero, false, false);
    // All D rows identical (A = ones) -> VGPR0 is valid for both lane halves:
    // lane<16: (M=0, N=lane); lane>=16: (M=8, N=lane-16). Both give pixel
    // hw + 4*(lane&15) + g, which is exactly this lane's store column.
    corr[g] = D[0] * ((A[0] > 0.1f) ? (1.0f / (float)CH) : 0.0f);
  }
  const v4f cv = {corr[0], corr[1], corr[2], corr[3]};

  // Broadcast to 81 output channels. Lane halves cover rows oc and oc+1 in
  // one full-wave 512B NT store (two 256B row segments per instruction).
  float* ob = out + (size_t)b * OUTC * HW + hw + 4 * col;
#pragma unroll
  for (int oc = 0; oc < OUTC - 1; oc += 2) {
    __builtin_nontemporal_store(cv, (v4f*)(ob + (size_t)(oc + half) * HW));
  }
  if (half == 0) {  // last row (oc = 80): half-wave store
    __builtin_nontemporal_store(cv, (v4f*)(ob + (size_t)(OUTC - 1) * HW));
  }
}

extern "C" void kernel_launch(void* const* d_in, const int* in_sizes, int n_in,
                              void* d_out, int out_size, void* d_ws, size_t ws_size,
                              hipStream_t stream) {
  const float* f1 = (const float*)d_in[0];
  const float* f2 = (const float*)d_in[1];
  float* out = (float*)d_out;

  // 4*192*320 = 245760 pixels; 64 pixels per wave; 8 waves per 256-thread block.
  const int total_waves = (BATCH * HW) / 64;        // 3840
  const int blocks      = (total_waves * 32) / 256; // 480
  hipLaunchKernelGGL(spike_corr_kernel, dim3(blocks), dim3(256), 0, stream,
                     f1, f2, out);
}